// PolicyValueNet_13735305413336
// MI455X (gfx1250) — compile-verified
//
#include <hip/hip_runtime.h>
#include <hip/hip_bf16.h>

typedef __attribute__((ext_vector_type(2))) float v2f;
typedef __attribute__((ext_vector_type(8))) float v8f;

#define N_NODES   32768
#define N_PER     512
#define E_EDGES   524288
#define E_PER     8192
#define NGRAPH    64
#define GNN_IN    64
#define EMB       128
#define HID       128
#define NA        4
#define EA        2
#define MAX_A     18432      // 512*4 + 8192*2
#define NODE_COLS 2048       // 512*4

static __device__ __forceinline__ v2f ld2(const float* p) {
    return *(const v2f*)p;   // callers guarantee 8B alignment (even float offset)
}

// ---------------------------------------------------------------------------
// Weight transpose: W [K,128] row-major -> WT [128,K] so B-fragments are
// contiguous b64 loads. Runs once per call, ~hundreds of KB total.
// ---------------------------------------------------------------------------
template <int K>
__global__ void transpose_w(const float* __restrict__ W, float* __restrict__ WT) {
    int idx = blockIdx.x * 256 + threadIdx.x;      // K*128 elements
    if (idx < K * EMB) {
        int k = idx >> 7;
        int c = idx & 127;
        WT[c * K + k] = W[idx];
    }
}

// ---------------------------------------------------------------------------
// WMMA f32 GEMM: C[M,128] = A[M,K] * W[K,128]; one wave per 32-row stripe.
// A-frag (16x4): lane<16 -> row=lane%16, K=kk+{0,1}; lane>=16 -> K=kk+{2,3}.
// B-frag from WT: one b64 per tile per K-step, reused by both row groups.
// C/D: vgpr j -> M = j + 8*(lane/16), N = lane%16.
// ---------------------------------------------------------------------------
template <int K>
__global__ __launch_bounds__(32) void gemm_nk(const float* __restrict__ A,
                                              const float* __restrict__ WT,
                                              float* __restrict__ C) {
    const int tile = blockIdx.x;                   // M/32 stripes
    const int lane = threadIdx.x;
    const int half = lane >> 4;
    const int l16  = lane & 15;
    const int k0   = half * 2;
    const int r0   = tile * 32 + l16;              // rows r0 and r0+16

    v8f acc[2][8];
#pragma unroll
    for (int g = 0; g < 2; ++g)
#pragma unroll
        for (int t = 0; t < 8; ++t) acc[g][t] = v8f{};

    for (int kk = 0; kk < K; kk += 4) {
        const int k = kk + k0;
        v2f a0 = ld2(A + r0 * K + k);
        v2f a1 = ld2(A + (r0 + 16) * K + k);
#pragma unroll
        for (int t = 0; t < 8; ++t) {
            v2f b = ld2(WT + (t * 16 + l16) * K + k);
            acc[0][t] = __builtin_amdgcn_wmma_f32_16x16x4_f32(
                false, a0, false, b, (short)0, acc[0][t], false, false);
            acc[1][t] = __builtin_amdgcn_wmma_f32_16x16x4_f32(
                false, a1, false, b, (short)0, acc[1][t], false, false);
        }
    }
#pragma unroll
    for (int g = 0; g < 2; ++g)
#pragma unroll
        for (int t = 0; t < 8; ++t)
#pragma unroll
            for (int j = 0; j < 8; ++j) {
                const int m = tile * 32 + g * 16 + j + half * 8;
                const int n = t * 16 + l16;
                C[m * EMB + n] = acc[g][t][j];
            }
}

// ------------------------- degree / normalization --------------------------
__global__ void deg_init(float* __restrict__ deg) {
    int n = blockIdx.x * 256 + threadIdx.x;
    if (n < N_NODES) deg[n] = 1.0f;                // self loop
}
__global__ void deg_edges(const int* __restrict__ dst, float* __restrict__ deg) {
    int e = blockIdx.x * 256 + threadIdx.x;
    if (e < E_EDGES) atomicAdd(&deg[dst[e]], 1.0f);
}
__global__ void deg_inv(float* __restrict__ deg) {
    int n = blockIdx.x * 256 + threadIdx.x;
    if (n < N_NODES) deg[n] = rsqrtf(deg[n]);      // deg >= 1 always
}

// ------------------------- GCN aggregation (float4) ------------------------
// out[n,:] = hW[n,:]*inv[n]^2 + bias   (self-loop + bias), 4 ch per thread
__global__ void agg_init(const float* __restrict__ hW, const float* __restrict__ inv,
                         const float* __restrict__ bias, float* __restrict__ out) {
    int idx = blockIdx.x * 256 + threadIdx.x;      // N*32
    int n = idx >> 5, q = idx & 31;
    float iv = inv[n];
    float s  = iv * iv;
    float4 h = ((const float4*)hW)[idx];
    float4 b = ((const float4*)bias)[q];
    float4 o = {h.x * s + b.x, h.y * s + b.y, h.z * s + b.z, h.w * s + b.w};
    ((float4*)out)[idx] = o;
}
// out[dst,:] += hW[src,:]*inv[src]*inv[dst], 4 ch per thread
__global__ void agg_edges(const float* __restrict__ hW, const int* __restrict__ src,
                          const int* __restrict__ dst, const float* __restrict__ inv,
                          float* __restrict__ out) {
    int idx = blockIdx.x * 256 + threadIdx.x;      // E*32
    int e = idx >> 5, q = idx & 31;
    int s = src[e], d = dst[e];
    float w = inv[s] * inv[d];
    float4 h = ((const float4*)(hW + s * EMB))[q];
    float* o = out + d * EMB + q * 4;
    atomicAdd(o + 0, h.x * w);
    atomicAdd(o + 1, h.y * w);
    atomicAdd(o + 2, h.z * w);
    atomicAdd(o + 3, h.w * w);
}
__global__ void relu_k(float* __restrict__ x, int n4) {
    int i = blockIdx.x * 256 + threadIdx.x;
    if (i < n4) {
        float4 v = ((const float4*)x)[i];
        float4 o = {fmaxf(v.x, 0.f), fmaxf(v.y, 0.f), fmaxf(v.z, 0.f), fmaxf(v.w, 0.f)};
        ((float4*)x)[i] = o;
    }
}

// ------------------------- node head (fused, 32 rows/wave) -----------------
__global__ __launch_bounds__(32) void node_head(const float* __restrict__ h,
                                                const float* __restrict__ wn1T,
                                                const float* __restrict__ bn1,
                                                const float* __restrict__ wn2,
                                                const float* __restrict__ bn2,
                                                float* __restrict__ out) {
    __shared__ float smem[32][HID];
    const int tile = blockIdx.x;                   // N/32
    const int lane = threadIdx.x;
    const int half = lane >> 4, l16 = lane & 15;
    const int k0   = half * 2;
    const int r0   = tile * 32 + l16;

    v8f acc[2][8];
#pragma unroll
    for (int g = 0; g < 2; ++g)
#pragma unroll
        for (int t = 0; t < 8; ++t) acc[g][t] = v8f{};

    for (int kk = 0; kk < EMB; kk += 4) {
        const int k = kk + k0;
        v2f a0 = ld2(h + r0 * EMB + k);
        v2f a1 = ld2(h + (r0 + 16) * EMB + k);
#pragma unroll
        for (int t = 0; t < 8; ++t) {
            v2f b = ld2(wn1T + (t * 16 + l16) * EMB + k);
            acc[0][t] = __builtin_amdgcn_wmma_f32_16x16x4_f32(
                false, a0, false, b, (short)0, acc[0][t], false, false);
            acc[1][t] = __builtin_amdgcn_wmma_f32_16x16x4_f32(
                false, a1, false, b, (short)0, acc[1][t], false, false);
        }
    }
#pragma unroll
    for (int g = 0; g < 2; ++g)
#pragma unroll
        for (int t = 0; t < 8; ++t)
#pragma unroll
            for (int j = 0; j < 8; ++j) {
                const int m = g * 16 + j + half * 8;
                const int n = t * 16 + l16;
                smem[m][n] = fmaxf(acc[g][t][j] + bn1[n], 0.0f);
            }
    __syncthreads();
    // 32 nodes x 4 actions = 128 outputs; 4 per lane
    for (int idx = lane; idx < 32 * NA; idx += 32) {
        const int r = idx >> 2, a = idx & 3;
        float s = bn2[a];
#pragma unroll 8
        for (int c = 0; c < HID; ++c) s += smem[r][c] * wn2[c * NA + a];
        const int node  = tile * 32 + r;
        const int g     = node / N_PER;
        const int local = node % N_PER;
        out[g * MAX_A + local * NA + a] = s;
    }
}

// ------------------------- edge head (fused gather, 32 rows/wave) ----------
__global__ __launch_bounds__(32) void edge_head(const float* __restrict__ h,
                                                const int* __restrict__ src,
                                                const int* __restrict__ dst,
                                                const float* __restrict__ we1T,
                                                const float* __restrict__ be1,
                                                const float* __restrict__ we2,
                                                const float* __restrict__ be2,
                                                float* __restrict__ out) {
    __shared__ float smem[32][HID];
    const int tile = blockIdx.x;                   // E/32
    const int lane = threadIdx.x;
    const int half = lane >> 4, l16 = lane & 15;
    const int k0   = half * 2;
    const int e0   = tile * 32 + l16;              // edges e0 and e0+16

    const float* hs0 = h + (long)src[e0] * EMB;
    const float* hd0 = h + (long)dst[e0] * EMB;
    const float* hs1 = h + (long)src[e0 + 16] * EMB;
    const float* hd1 = h + (long)dst[e0 + 16] * EMB;

    v8f acc[2][8];
#pragma unroll
    for (int g = 0; g < 2; ++g)
#pragma unroll
        for (int t = 0; t < 8; ++t) acc[g][t] = v8f{};

    for (int kk = 0; kk < 2 * EMB; kk += 4) {
        const int k = kk + k0;                     // k,k+1 always same side of 128
        const float* b0 = (k < EMB) ? hs0 : (hd0 - EMB);
        const float* b1 = (k < EMB) ? hs1 : (hd1 - EMB);
        v2f a0 = ld2(b0 + k);
        v2f a1 = ld2(b1 + k);
#pragma unroll
        for (int t = 0; t < 8; ++t) {
            v2f b = ld2(we1T + (t * 16 + l16) * (2 * EMB) + k);
            acc[0][t] = __builtin_amdgcn_wmma_f32_16x16x4_f32(
                false, a0, false, b, (short)0, acc[0][t], false, false);
            acc[1][t] = __builtin_amdgcn_wmma_f32_16x16x4_f32(
                false, a1, false, b, (short)0, acc[1][t], false, false);
        }
    }
#pragma unroll
    for (int g = 0; g < 2; ++g)
#pragma unroll
        for (int t = 0; t < 8; ++t)
#pragma unroll
            for (int j = 0; j < 8; ++j) {
                const int m = g * 16 + j + half * 8;
                const int n = t * 16 + l16;
                smem[m][n] = fmaxf(acc[g][t][j] + be1[n], 0.0f);
            }
    __syncthreads();
    // 32 edges x 2 actions = 64 outputs; 2 per lane
    for (int idx = lane; idx < 32 * EA; idx += 32) {
        const int r = idx >> 1, a = idx & 1;
        float s = be2[a];
#pragma unroll 8
        for (int c = 0; c < HID; ++c) s += smem[r][c] * we2[c * EA + a];
        const int e     = tile * 32 + r;
        const int g     = e / E_PER;
        const int local = e % E_PER;
        out[g * MAX_A + NODE_COLS + local * EA + a] = s;
    }
}

// ------------------------- pooling + value head ----------------------------
__global__ void pool_mean(const float* __restrict__ h, float* __restrict__ g) {
    const int b = blockIdx.x, c = threadIdx.x;     // 128 threads
    float s = 0.0f;
    for (int n = 0; n < N_PER; ++n) s += h[(b * N_PER + n) * EMB + c];
    g[b * EMB + c] = s * (1.0f / N_PER);
}

__global__ void value_head(const float* __restrict__ g, const float* __restrict__ wv1,
                           const float* __restrict__ bv1, const float* __restrict__ wv2,
                           const float* __restrict__ bv2, float* __restrict__ out) {
    __shared__ float t[HID];
    const int b = blockIdx.x, j = threadIdx.x;     // 128 threads
    float s = bv1[j];
    for (int c = 0; c < EMB; ++c) s += g[b * EMB + c] * wv1[c * HID + j];
    t[j] = fmaxf(s, 0.0f);
    __syncthreads();
    if (j == 0) {
        float v = bv2[0];
        for (int k = 0; k < HID; ++k) v += t[k] * wv2[k];
        out[NGRAPH * MAX_A + b] = v;
    }
}

// ---------------------------------------------------------------------------
extern "C" void kernel_launch(void* const* d_in, const int* in_sizes, int n_in,
                              void* d_out, int out_size, void* d_ws, size_t ws_size,
                              hipStream_t stream) {
    const float* x    = (const float*)d_in[0];
    const int* ei     = (const int*)d_in[1];
    const int* src    = ei;
    const int* dst    = ei + E_EDGES;
    const float* w_g1 = (const float*)d_in[5];
    const float* b_g1 = (const float*)d_in[6];
    const float* w_g2 = (const float*)d_in[7];
    const float* b_g2 = (const float*)d_in[8];
    const float* w_g3 = (const float*)d_in[9];
    const float* b_g3 = (const float*)d_in[10];
    const float* wn1  = (const float*)d_in[11];
    const float* bn1  = (const float*)d_in[12];
    const float* wn2  = (const float*)d_in[13];
    const float* bn2  = (const float*)d_in[14];
    const float* we1  = (const float*)d_in[15];
    const float* be1  = (const float*)d_in[16];
    const float* we2  = (const float*)d_in[17];
    const float* be2  = (const float*)d_in[18];
    const float* wv1  = (const float*)d_in[19];
    const float* bv1  = (const float*)d_in[20];
    const float* wv2  = (const float*)d_in[21];
    const float* bv2  = (const float*)d_in[22];
    float* out = (float*)d_out;

    // workspace layout
    float* deg   = (float*)d_ws;                   // N
    float* hbuf  = deg + N_NODES;                  // N*EMB  (h, post-ReLU)
    float* hW    = hbuf + (size_t)N_NODES * EMB;   // N*EMB  (h @ W scratch)
    float* gbuf  = hW + (size_t)N_NODES * EMB;     // B*EMB
    float* wg1T  = gbuf + NGRAPH * EMB;            // 128*64
    float* wg2T  = wg1T + GNN_IN * EMB;            // 128*128
    float* wg3T  = wg2T + EMB * EMB;               // 128*128
    float* wn1T  = wg3T + EMB * EMB;               // 128*128
    float* we1T  = wn1T + EMB * EMB;               // 128*256

    const int NE4 = N_NODES * EMB / 4;             // 1M float4
    // transposed weights for coalesced b64 B-fragment loads
    transpose_w<GNN_IN><<<(GNN_IN * EMB + 255) / 256, 256, 0, stream>>>(w_g1, wg1T);
    transpose_w<EMB><<<(EMB * EMB + 255) / 256, 256, 0, stream>>>(w_g2, wg2T);
    transpose_w<EMB><<<(EMB * EMB + 255) / 256, 256, 0, stream>>>(w_g3, wg3T);
    transpose_w<EMB><<<(EMB * EMB + 255) / 256, 256, 0, stream>>>(wn1, wn1T);
    transpose_w<2 * EMB><<<(2 * EMB * EMB + 255) / 256, 256, 0, stream>>>(we1, we1T);

    // symmetric normalization (once; same for all 3 layers)
    deg_init<<<(N_NODES + 255) / 256, 256, 0, stream>>>(deg);
    deg_edges<<<(E_EDGES + 255) / 256, 256, 0, stream>>>(dst, deg);
    deg_inv<<<(N_NODES + 255) / 256, 256, 0, stream>>>(deg);

    // layer 1: x [N,64] -> hbuf [N,128]
    gemm_nk<GNN_IN><<<N_NODES / 32, 32, 0, stream>>>(x, wg1T, hW);
    agg_init<<<N_NODES * 32 / 256, 256, 0, stream>>>(hW, deg, b_g1, hbuf);
    agg_edges<<<E_EDGES * 32 / 256, 256, 0, stream>>>(hW, src, dst, deg, hbuf);
    relu_k<<<(NE4 + 255) / 256, 256, 0, stream>>>(hbuf, NE4);

    // layer 2
    gemm_nk<EMB><<<N_NODES / 32, 32, 0, stream>>>(hbuf, wg2T, hW);
    agg_init<<<N_NODES * 32 / 256, 256, 0, stream>>>(hW, deg, b_g2, hbuf);
    agg_edges<<<E_EDGES * 32 / 256, 256, 0, stream>>>(hW, src, dst, deg, hbuf);
    relu_k<<<(NE4 + 255) / 256, 256, 0, stream>>>(hbuf, NE4);

    // layer 3
    gemm_nk<EMB><<<N_NODES / 32, 32, 0, stream>>>(hbuf, wg3T, hW);
    agg_init<<<N_NODES * 32 / 256, 256, 0, stream>>>(hW, deg, b_g3, hbuf);
    agg_edges<<<E_EDGES * 32 / 256, 256, 0, stream>>>(hW, src, dst, deg, hbuf);
    relu_k<<<(NE4 + 255) / 256, 256, 0, stream>>>(hbuf, NE4);

    // pooling + heads (logits columns fully covered: [0,2048) nodes, [2048,18432) edges)
    pool_mean<<<NGRAPH, EMB, 0, stream>>>(hbuf, gbuf);
    node_head<<<N_NODES / 32, 32, 0, stream>>>(hbuf, wn1T, bn1, wn2, bn2, out);
    edge_head<<<E_EDGES / 32, 32, 0, stream>>>(hbuf, src, dst, we1T, be1, we2, be2, out);
    value_head<<<NGRAPH, EMB, 0, stream>>>(gbuf, wv1, bv1, wv2, bv2, out);
}